// DistSparseMoe_38972533244544
// MI455X (gfx1250) — compile-verified
//
#include <hip/hip_runtime.h>

typedef float v2f __attribute__((ext_vector_type(2)));
typedef float v8f __attribute__((ext_vector_type(8)));

#define T_TOKENS 16384
#define H_DIM    2048
#define E_NUM    8

// ---------------- Kernel 1: router (logits -> softmax -> top-1) ----------------
// One wave per token. Lane-parallel over H with 8 expert accumulators.
__global__ __launch_bounds__(256) void moe_router(const float* __restrict__ x,
                                                  const float* __restrict__ Wg,
                                                  const float* __restrict__ bg,
                                                  int* __restrict__ best,
                                                  float* __restrict__ bp) {
    const int lane = threadIdx.x & 31;
    const int wid  = threadIdx.x >> 5;
    const int t    = blockIdx.x * 8 + wid;
    const float* xr = x + (size_t)t * H_DIM;

    float acc[E_NUM];
#pragma unroll
    for (int e = 0; e < E_NUM; ++e) acc[e] = 0.0f;

    for (int i = lane; i < H_DIM; i += 32) {
        float xv = xr[i];
        const float4* wr = (const float4*)(Wg + (size_t)i * E_NUM);
        float4 lo = wr[0];
        float4 hi = wr[1];
        acc[0] += xv * lo.x; acc[1] += xv * lo.y;
        acc[2] += xv * lo.z; acc[3] += xv * lo.w;
        acc[4] += xv * hi.x; acc[5] += xv * hi.y;
        acc[6] += xv * hi.z; acc[7] += xv * hi.w;
    }
#pragma unroll
    for (int off = 16; off >= 1; off >>= 1) {
#pragma unroll
        for (int e = 0; e < E_NUM; ++e) acc[e] += __shfl_xor(acc[e], off, 32);
    }
    if (lane == 0) {
        float l[E_NUM];
#pragma unroll
        for (int e = 0; e < E_NUM; ++e) l[e] = acc[e] + bg[e];
        int bi = 0;
        float m = l[0];
#pragma unroll
        for (int e = 1; e < E_NUM; ++e) {
            if (l[e] > m) { m = l[e]; bi = e; }   // strict > keeps first max (jnp.argmax)
        }
        float s = 0.0f;
#pragma unroll
        for (int e = 0; e < E_NUM; ++e) s += __expf(l[e] - m);
        best[t] = bi;
        bp[t]   = 1.0f / s;                        // exp(0)/sum
    }
}

// ---------------- Kernel 2a: per-32-token expert histogram ----------------
__global__ __launch_bounds__(32) void moe_hist(const int* __restrict__ best,
                                               int* __restrict__ counts) {
    const int lane = threadIdx.x;
    const int t = blockIdx.x * 32 + lane;
    const int e = best[t];
#pragma unroll
    for (int ee = 0; ee < E_NUM; ++ee) {
        unsigned long long m = __ballot(e == ee);
        if (lane == 0) counts[blockIdx.x * E_NUM + ee] = __popcll(m);
    }
}

// ---------------- Kernel 2b: scan histograms -> stable scatter offsets ----------------
__global__ __launch_bounds__(32) void moe_scan(const int* __restrict__ counts,
                                               int* __restrict__ blockoff) {
    __shared__ int tot[E_NUM];
    __shared__ int ebase[E_NUM];
    const int e = threadIdx.x;
    if (e < E_NUM) {
        int run = 0;
        for (int b = 0; b < T_TOKENS / 32; ++b) {
            blockoff[b * E_NUM + e] = run;
            run += counts[b * E_NUM + e];
        }
        tot[e] = run;
    }
    __syncthreads();
    if (threadIdx.x == 0) {
        int base = 0;
        for (int i = 0; i < E_NUM; ++i) { ebase[i] = base; base += tot[i]; }
    }
    __syncthreads();
    if (e < E_NUM) {
        const int eb = ebase[e];
        for (int b = 0; b < T_TOKENS / 32; ++b) blockoff[b * E_NUM + e] += eb;
    }
}

// ---------------- Kernel 2c: stable scatter -> order[] ----------------
__global__ __launch_bounds__(32) void moe_scatter(const int* __restrict__ best,
                                                  const int* __restrict__ blockoff,
                                                  int* __restrict__ order) {
    const int lane = threadIdx.x;
    const int t = blockIdx.x * 32 + lane;
    const int e = best[t];
    unsigned long long mymask = 0ull;
#pragma unroll
    for (int ee = 0; ee < E_NUM; ++ee) {
        unsigned long long m = __ballot(e == ee);
        if (e == ee) mymask = m;
    }
    const unsigned long long lt = ((unsigned long long)1 << lane) - 1ull;
    const int rank = __popcll(mymask & lt);
    const int pos = blockoff[blockIdx.x * E_NUM + e] + rank;
    order[pos] = t;
}

// ---------------- Kernel 3: gathered expert GEMM via V_WMMA_F32_16X16X4_F32 ----------------
#define BM 128
#define BN 128
#define TK 32
#define ASTRIDE 36     // A: padded row stride, 144 B (16B aligned, conflict-free frag reads)
#define BPSTRIDE 144   // B: pair-row stride in float2 units; kp parity flips bank half

__global__ __launch_bounds__(256) void moe_expert_gemm(const float* __restrict__ X,
                                                       const float* __restrict__ We,
                                                       const float* __restrict__ be,
                                                       const int* __restrict__ order,
                                                       const float* __restrict__ bp,
                                                       float* __restrict__ out) {
    __shared__ float As[BM * ASTRIDE];                  // 18432 B
    __shared__ float Bs[(TK / 2) * BPSTRIDE * 2];       // 18432 B, K-pair interleaved
    __shared__ int   ord_s[BM];
    __shared__ float bp_s[BM];

    const int tid  = threadIdx.x;
    const int lane = tid & 31;
    const int wid  = tid >> 5;
    const int wave_m = wid >> 1;   // 0..3 : 32 output rows each
    const int wave_n = wid & 1;    // 0..1 : 64 output cols each
    const int bm = blockIdx.y * BM;
    const int bn = blockIdx.x * BN;

    if (tid < BM) {
        const int g = bm + tid;
        ord_s[tid] = order[g];
        bp_s[tid]  = bp[g];        // positional scale (reference never un-permutes)
    }
    __syncthreads();

    v8f acc[2][4];
#pragma unroll
    for (int i = 0; i < 2; ++i)
#pragma unroll
        for (int j = 0; j < 4; ++j) acc[i][j] = v8f{};

    const int lhalf = lane >> 4;   // 0: lanes hold (K,K+1)   1: (K+2,K+3)
    const int lmod  = lane & 15;

    // Per-thread staging coordinates (loop-invariant)
    const int a_row = tid >> 1;                 // 0..127  (A: 2 threads per row, 4 float4 each)
    const int a_c0  = (tid & 1) << 4;           // 0 or 16

    for (int k0 = 0; k0 < H_DIM; k0 += TK) {
        // ---- Stage A tile (rows gathered through order), 128x32 fp32 ----
        {
            const int grow = ord_s[a_row];
            const float* src = X + (size_t)grow * H_DIM + k0 + a_c0;
            float* dst = &As[a_row * ASTRIDE + a_c0];
            *(float4*)(dst)     = *(const float4*)(src);
            *(float4*)(dst + 4) = *(const float4*)(src + 4);
            *(float4*)(dst + 8) = *(const float4*)(src + 8);
            *(float4*)(dst + 12)= *(const float4*)(src + 12);
            if (k0 + TK < H_DIM) {
                __builtin_prefetch(src + TK, 0, 3);          // next A stage
                __builtin_prefetch(src + TK + 8, 0, 3);
            }
        }
        // ---- Stage B tile, 32x128 fp32, K-pair interleaved:
        //      Bs2[kp][n] = { We[k0+2kp][bn+n], We[k0+2kp+1][bn+n] } ----
#pragma unroll
        for (int p = 0; p < 2; ++p) {
            const int slot = p * 256 + tid;     // 0..511 pair-row float4 slots
            const int rp  = slot >> 5;          // 0..15 (k-pair row)
            const int c4  = (slot & 31) << 2;   // 0..124
            const float* s0 = We + (size_t)(k0 + 2 * rp)     * H_DIM + bn + c4;
            const float* s1 = We + (size_t)(k0 + 2 * rp + 1) * H_DIM + bn + c4;
            const float4 ra = *(const float4*)s0;
            const float4 rb = *(const float4*)s1;
            float4 p0 = make_float4(ra.x, rb.x, ra.y, rb.y);
            float4 p1 = make_float4(ra.z, rb.z, ra.w, rb.w);
            float* dst = &Bs[(rp * BPSTRIDE + c4) * 2];
            *(float4*)(dst)     = p0;
            *(float4*)(dst + 4) = p1;
            if (k0 + TK < H_DIM) {
                __builtin_prefetch(s0 + (size_t)TK * H_DIM, 0, 3);   // next B stage
            }
        }
        __syncthreads();

#pragma unroll
        for (int kk = 0; kk < TK; kk += 4) {
            v2f a[2], b[4];
#pragma unroll
            for (int im = 0; im < 2; ++im) {
                const int row = wave_m * 32 + im * 16 + lmod;
                const int col = kk + 2 * lhalf;
                a[im] = *(const v2f*)&As[row * ASTRIDE + col];
            }
            const int kp = (kk >> 1) + lhalf;   // pair row; parity -> disjoint bank half
#pragma unroll
            for (int in = 0; in < 4; ++in) {
                const int col = wave_n * 64 + in * 16 + lmod;
                b[in] = *(const v2f*)&Bs[(kp * BPSTRIDE + col) * 2];
            }
#pragma unroll
            for (int im = 0; im < 2; ++im)
#pragma unroll
                for (int in = 0; in < 4; ++in)
                    acc[im][in] = __builtin_amdgcn_wmma_f32_16x16x4_f32(
                        false, a[im], false, b[in], (short)0, acc[im][in], false, false);
        }
        __syncthreads();
    }

    // Epilogue: out[row] = (acc + be[col]) * bp[row]  (positional scale)
#pragma unroll
    for (int im = 0; im < 2; ++im) {
        const int lrow_base = wave_m * 32 + im * 16 + 8 * lhalf;
#pragma unroll
        for (int in = 0; in < 4; ++in) {
            const int gcol = bn + wave_n * 64 + in * 16 + lmod;
            const float bias = be[gcol];
#pragma unroll
            for (int r = 0; r < 8; ++r) {
                const int lrow = lrow_base + r;
                out[(size_t)(bm + lrow) * H_DIM + gcol] =
                    (acc[im][in][r] + bias) * bp_s[lrow];
            }
        }
    }
}

extern "C" void kernel_launch(void* const* d_in, const int* in_sizes, int n_in,
                              void* d_out, int out_size, void* d_ws, size_t ws_size,
                              hipStream_t stream) {
    const float* x  = (const float*)d_in[0];
    const float* Wg = (const float*)d_in[1];
    const float* bg = (const float*)d_in[2];
    const float* We = (const float*)d_in[3];
    const float* be = (const float*)d_in[4];
    float* out = (float*)d_out;

    char* ws = (char*)d_ws;
    int*   best     = (int*)  (ws);
    float* bp       = (float*)(ws + 1 * (size_t)T_TOKENS * 4);
    int*   order    = (int*)  (ws + 2 * (size_t)T_TOKENS * 4);
    int*   counts   = (int*)  (ws + 3 * (size_t)T_TOKENS * 4);
    int*   blockoff = (int*)  (ws + 3 * (size_t)T_TOKENS * 4 + (T_TOKENS / 32) * E_NUM * 4);

    moe_router <<<T_TOKENS / 8, 256, 0, stream>>>(x, Wg, bg, best, bp);
    moe_hist   <<<T_TOKENS / 32, 32, 0, stream>>>(best, counts);
    moe_scan   <<<1, 32, 0, stream>>>(counts, blockoff);
    moe_scatter<<<T_TOKENS / 32, 32, 0, stream>>>(best, blockoff, order);

    dim3 grid(H_DIM / BN, T_TOKENS / BM);
    moe_expert_gemm<<<grid, 256, 0, stream>>>(x, We, be, order, bp, out);
}